// CRF_58815282152109
// MI455X (gfx1250) — compile-verified
//
#include <hip/hip_runtime.h>
#include <math.h>

// CRF forward: B=256, S=1024, T=128, START=126, END=127
static constexpr int Bn = 256;
static constexpr int Sn = 1024;
static constexpr int Tn = 128;
static constexpr int STARTn = 126;
static constexpr int ENDn = 127;
static constexpr float FINF_MIN = -10000.0f;

typedef __attribute__((ext_vector_type(2))) float v2f;
typedef __attribute__((ext_vector_type(8))) float v8f;

// ---------------------------------------------------------------------------
// Prep kernel: Wg[k*Tn + i] = exp(trans[i*Tn + k])   (exp-space transition,
// transposed so the recurrence is a plain GEMM  alphaP @ Wg ),
// Lg[i] = log(1 + sum_{j != START} exp(trans[i*Tn + j]))  (exact step-0 value)
// ---------------------------------------------------------------------------
__global__ void crf_prep_kernel(const float* __restrict__ trans,
                                float* __restrict__ Wg,
                                float* __restrict__ Lg) {
  int i = threadIdx.x;  // 0..127 : output tag index
  float suml = 0.0f;
  for (int j = 0; j < Tn; ++j) {
    float e = __expf(trans[i * Tn + j]);   // -10000 -> 0 (true structural zero)
    Wg[j * Tn + i] = e;
    if (j != STARTn) suml += e;
  }
  Lg[i] = __logf(1.0f + suml);
}

// ---------------------------------------------------------------------------
// Main kernel: 16 blocks x 256 threads (8 waves).
// Block bx handles batch rows [16*bx, 16*bx+16).
// Wave w handles output columns [16*w, 16*w+16); its 128x16 slab of Wg stays
// resident in 64 VGPRs (32 x v2f WMMA B-fragments) for the whole s-loop.
// ---------------------------------------------------------------------------
__global__ __launch_bounds__(256) void crf_main_kernel(
    const float* __restrict__ feats, const int* __restrict__ blen,
    const float* __restrict__ trans, const float* __restrict__ Wg,
    const float* __restrict__ Lg, float* __restrict__ out) {
  __shared__ __align__(16) float s_alpha[16][132];  // padded: 132 mod 64 banks
  __shared__ __align__(16) float s_ea[16][132];     // exp(alpha - m)
  __shared__ float s_m[16];
  __shared__ int s_len[16];

  const int tid = threadIdx.x;
  const int wave = tid >> 5;
  const int lane = tid & 31;
  const int hi = lane >> 4;   // half-wave select (wave32: two groups of 16)
  const int ln = lane & 15;
  const int n0 = wave * 16;   // this wave's output-column tile
  const int b0 = blockIdx.x * 16;
  const int row = tid >> 4;   // reduction-stage row (16 threads per row)
  const int cl = tid & 15;

  if (tid < 16) s_len[tid] = blen[b0 + tid];

  // ---- step 0 (always alive since batch_len >= 1), computed analytically ----
  {
    const float* fr = feats + (size_t)(b0 + row) * Sn * Tn;  // s = 0
#pragma unroll
    for (int q = 0; q < 8; ++q) {
      int c = cl + 16 * q;
      s_alpha[row][c] = fr[c] + FINF_MIN + Lg[c];
    }
  }

  // ---- load resident W B-fragments: lane ln -> column n0+ln,
  // fragment kk covers K = 4*kk .. 4*kk+3 ; VGPR pair = rows (K0+2*hi, K0+2*hi+1)
  v2f wb[32];
#pragma unroll
  for (int kk = 0; kk < 32; ++kk) {
    int k0 = 4 * kk + 2 * hi;
    int n = n0 + ln;
    wb[kk].x = Wg[k0 * Tn + n];
    wb[kk].y = Wg[(k0 + 1) * Tn + n];
  }

  // ---- prefetch feats for s = 1 (mapping matches the C-matrix writeback) ----
  float f[8];
#pragma unroll
  for (int r = 0; r < 8; ++r)
    f[r] = feats[((size_t)(b0 + r + 8 * hi) * Sn + 1) * Tn + n0 + ln];

  // ---- sequential recurrence over s ----
  for (int s = 1; s < Sn; ++s) {
    __syncthreads();  // alpha from previous step visible

    // per-batch-row max + exp into s_ea  (16 threads per row, shfl width 16)
    float av[8];
    float mx = -INFINITY;
#pragma unroll
    for (int q = 0; q < 8; ++q) {
      av[q] = s_alpha[row][cl + 16 * q];
      mx = fmaxf(mx, av[q]);
    }
#pragma unroll
    for (int off = 8; off >= 1; off >>= 1)
      mx = fmaxf(mx, __shfl_xor(mx, off, 16));
    if (cl == 0) s_m[row] = mx;
#pragma unroll
    for (int q = 0; q < 8; ++q)
      s_ea[row][cl + 16 * q] = __expf(av[q] - mx);

    __syncthreads();  // s_ea / s_m ready

    // GEMM tile: C[16 batches x 16 cols] = EA[16 x 128] * W[128 x 16]
    v8f c = {};
#pragma unroll
    for (int kk = 0; kk < 32; ++kk) {
      // A fragment: lane ln = batch row M, K pair (4kk+2hi, 4kk+2hi+1)
      v2f a = *(const v2f*)&s_ea[ln][4 * kk + 2 * hi];
      c = __builtin_amdgcn_wmma_f32_16x16x4_f32(
          /*neg_a=*/false, a, /*neg_b=*/false, wb[kk],
          /*c_mod=*/(short)0, c, /*reuse_a=*/false, /*reuse_b=*/false);
    }

    // alpha writeback: VGPR r -> batch row M = r + 8*hi, col = n0 + ln
#pragma unroll
    for (int r = 0; r < 8; ++r) {
      int M = r + 8 * hi;
      if (s < s_len[M])
        s_alpha[M][n0 + ln] = f[r] + __logf(c[r]) + s_m[M];
    }

    // prefetch next step's feats (overlaps with next iteration's reduction)
    int sn = (s + 1 < Sn) ? (s + 1) : (Sn - 1);
#pragma unroll
    for (int r = 0; r < 8; ++r)
      f[r] = feats[((size_t)(b0 + r + 8 * hi) * Sn + sn) * Tn + n0 + ln];
  }

  __syncthreads();

  // ---- final: out[b] = logsumexp_i( alpha[b,i] + trans[END,i] ) ----
  {
    float tv[8];
    float mx = -INFINITY;
#pragma unroll
    for (int q = 0; q < 8; ++q) {
      int cc = cl + 16 * q;
      tv[q] = s_alpha[row][cc] + trans[ENDn * Tn + cc];
      mx = fmaxf(mx, tv[q]);
    }
#pragma unroll
    for (int off = 8; off >= 1; off >>= 1)
      mx = fmaxf(mx, __shfl_xor(mx, off, 16));
    float sum = 0.0f;
#pragma unroll
    for (int q = 0; q < 8; ++q) sum += __expf(tv[q] - mx);
#pragma unroll
    for (int off = 8; off >= 1; off >>= 1) sum += __shfl_xor(sum, off, 16);
    if (cl == 0) out[b0 + row] = mx + __logf(sum);
  }
}

// ---------------------------------------------------------------------------
extern "C" void kernel_launch(void* const* d_in, const int* in_sizes, int n_in,
                              void* d_out, int out_size, void* d_ws,
                              size_t ws_size, hipStream_t stream) {
  const float* feats = (const float*)d_in[0];   // [B,S,T] f32
  const int* blen = (const int*)d_in[1];        // [B] i32
  const float* trans = (const float*)d_in[2];   // [T,T] f32

  float* Wg = (float*)d_ws;          // Tn*Tn floats (64 KB)
  float* Lg = Wg + Tn * Tn;          // Tn floats

  crf_prep_kernel<<<1, Tn, 0, stream>>>(trans, Wg, Lg);
  crf_main_kernel<<<Bn / 16, 256, 0, stream>>>(feats, blen, trans, Wg, Lg,
                                               (float*)d_out);
}